// HierarchicalItem2Vec_88862873354427
// MI455X (gfx1250) — compile-verified
//
#include <hip/hip_runtime.h>
#include <hip/hip_bf16.h>
#include <math.h>

typedef __attribute__((ext_vector_type(2))) float v2f;
typedef __attribute__((ext_vector_type(8))) float v8f;

#define DIM     128
#define NKK     32          // 128 / 4 (K per V_WMMA_F32_16X16X4_F32)
#define LPATH   24
#define LCAT    6
#define BATCH   16384
#define NTILES  (BATCH/16)  // 1024 tiles of 16 rows
#define EPSV    1e-8f
#define LAMBDA_CAT 0.1f

// Extract acc[idx] (idx in 0..7) with a cndmask chain (no scratch, no EXEC games).
__device__ __forceinline__ float diag_select(v8f a, int idx) {
    float r = a[0];
    r = (idx == 1) ? a[1] : r;
    r = (idx == 2) ? a[2] : r;
    r = (idx == 3) ? a[3] : r;
    r = (idx == 4) ? a[4] : r;
    r = (idx == 5) ? a[5] : r;
    r = (idx == 6) ? a[6] : r;
    r = (idx == 7) ? a[7] : r;
    return r;
}

__global__ __launch_bounds__(128)
void hi2v_main(const int* __restrict__ center_ids, const int* __restrict__ context_ids,
               const float* __restrict__ item_emb, const float* __restrict__ cat_emb,
               const float* __restrict__ node_w, const int* __restrict__ path_nodes,
               const float* __restrict__ path_codes, const float* __restrict__ path_mask,
               const int* __restrict__ cat_path, const float* __restrict__ cat_mask,
               float* __restrict__ part)
{
    const int lane = threadIdx.x & 31;
    const int wave = threadIdx.x >> 5;
    const int tile = blockIdx.x * (blockDim.x >> 5) + wave;   // 16 batch rows / tile
    const int n    = lane & 15;      // row-in-tile this lane co-loads (A: M, B: N)
    const int half = lane >> 4;      // 0: K-pair {0,1}; 1: K-pair {2,3}
    const int row  = tile * 16 + n;

    const int cid = center_ids[row];
    const int ctx = context_ids[row];

    // ---- A fragments: 16 center vectors, fp32 WMMA layout, kept in VGPRs ----
    v2f areg[NKK];
    const float* cvec = item_emb + (size_t)cid * DIM;
    float cn2 = 0.f;
#pragma unroll
    for (int kk = 0; kk < NKK; ++kk) {
        areg[kk] = *(const v2f*)(cvec + 4 * kk + 2 * half);
        cn2 += areg[kk].x * areg[kk].x + areg[kk].y * areg[kk].y;
    }
    cn2 += __shfl_xor(cn2, 16, 32);               // combine complementary halves
    const float cn = fmaxf(sqrtf(cn2), EPSV);     // ||center|| for this lane's row

    // Diagonal (j,j) of the 16x16 f32 D tile: lane j holds it in VGPR j (j<8),
    // lane j+16 holds it in VGPR j-8 (j>=8)  -> active lanes {0..7, 24..31}.
    const bool  active = (lane < 8) || (lane >= 24);
    const float af     = active ? 1.f : 0.f;
    const int   didx   = lane & 7;

    // ---- hierarchical-softmax BCE over huffman path ----
    float hs_sum = 0.f, hs_cnt = 0.f;
    for (int l = 0; l < LPATH; ++l) {
        const int   node = path_nodes[ctx * LPATH + l];
        const float code = path_codes[ctx * LPATH + l];
        const float msk  = path_mask [ctx * LPATH + l];
        const float* wv  = node_w + (size_t)node * DIM;
        v8f acc = {0.f, 0.f, 0.f, 0.f, 0.f, 0.f, 0.f, 0.f};
#pragma unroll
        for (int kk = 0; kk < NKK; ++kk) {
            v2f b = *(const v2f*)(wv + 4 * kk + 2 * half);
            acc = __builtin_amdgcn_wmma_f32_16x16x4_f32(
                      false, areg[kk], false, b, (short)0, acc, false, false);
        }
        const float logit = diag_select(acc, didx);
        const float bce = fmaxf(logit, 0.f) - logit * code
                        + log1pf(expf(-fabsf(logit)));
        hs_sum += af * bce * msk;
        hs_cnt += af * msk;
    }

    // ---- cosine loss over category path ----
    float cat_per = 0.f, cat_cnt = 0.f;
    for (int l = 0; l < LCAT; ++l) {
        const int   cp = cat_path[cid * LCAT + l];
        const float cm = cat_mask[cid * LCAT + l];
        const float* ev = cat_emb + (size_t)cp * DIM;
        v8f acc = {0.f, 0.f, 0.f, 0.f, 0.f, 0.f, 0.f, 0.f};
        float en2 = 0.f;
#pragma unroll
        for (int kk = 0; kk < NKK; ++kk) {
            v2f b = *(const v2f*)(ev + 4 * kk + 2 * half);
            en2 += b.x * b.x + b.y * b.y;
            acc = __builtin_amdgcn_wmma_f32_16x16x4_f32(
                      false, areg[kk], false, b, (short)0, acc, false, false);
        }
        en2 += __shfl_xor(en2, 16, 32);
        const float en  = fmaxf(sqrtf(en2), EPSV);
        const float dot = diag_select(acc, didx);
        const float cosv = dot / (cn * en);
        cat_per += af * (1.f - cosv) * cm;
        cat_cnt += af * cm;
    }
    const float has   = (cat_cnt > 0.f) ? af : 0.f;
    const float pmean = (cat_cnt > 0.f) ? (cat_per / cat_cnt) : 0.f;

    // ---- wave reduction (inactive lanes carry zeros) -> per-tile partials ----
    float v0 = hs_sum, v1 = hs_cnt, v2 = af * pmean, v3 = has;
#pragma unroll
    for (int s = 16; s >= 1; s >>= 1) {
        v0 += __shfl_xor(v0, s, 32);
        v1 += __shfl_xor(v1, s, 32);
        v2 += __shfl_xor(v2, s, 32);
        v3 += __shfl_xor(v3, s, 32);
    }
    if (lane == 0) {
        part[tile]              = v0;
        part[NTILES     + tile] = v1;
        part[2 * NTILES + tile] = v2;
        part[3 * NTILES + tile] = v3;
    }
}

__global__ void hi2v_finalize(const float* __restrict__ part, float* __restrict__ out)
{
    __shared__ float s0[256], s1[256], s2[256], s3[256];
    const int t = threadIdx.x;
    float a = 0.f, b = 0.f, c = 0.f, d = 0.f;
    for (int i = t; i < NTILES; i += 256) {
        a += part[i];
        b += part[NTILES + i];
        c += part[2 * NTILES + i];
        d += part[3 * NTILES + i];
    }
    s0[t] = a; s1[t] = b; s2[t] = c; s3[t] = d;
    __syncthreads();
    for (int s = 128; s > 0; s >>= 1) {
        if (t < s) {
            s0[t] += s0[t + s];
            s1[t] += s1[t + s];
            s2[t] += s2[t + s];
            s3[t] += s3[t + s];
        }
        __syncthreads();
    }
    if (t == 0) {
        const float hs  = s0[0] / s1[0];
        const float cat = (s3[0] > 0.f) ? (s2[0] / s3[0]) : 0.f;
        out[0] = hs + LAMBDA_CAT * cat;
    }
}

extern "C" void kernel_launch(void* const* d_in, const int* in_sizes, int n_in,
                              void* d_out, int out_size, void* d_ws, size_t ws_size,
                              hipStream_t stream)
{
    const int*   center_ids  = (const int*)  d_in[0];
    const int*   context_ids = (const int*)  d_in[1];
    const float* item_emb    = (const float*)d_in[2];
    const float* cat_emb     = (const float*)d_in[3];
    const float* node_w      = (const float*)d_in[4];
    const int*   path_nodes  = (const int*)  d_in[5];
    const float* path_codes  = (const float*)d_in[6];
    const float* path_mask   = (const float*)d_in[7];
    const int*   cat_path    = (const int*)  d_in[8];
    const float* cat_mask    = (const float*)d_in[9];

    float* part = (float*)d_ws;            // 4 * NTILES floats, fully rewritten
    float* out  = (float*)d_out;

    // 4 waves per block, 1 tile per wave -> NTILES/4 blocks
    hi2v_main<<<NTILES / 4, 128, 0, stream>>>(
        center_ids, context_ids, item_emb, cat_emb, node_w,
        path_nodes, path_codes, path_mask, cat_path, cat_mask, part);
    hi2v_finalize<<<1, 256, 0, stream>>>(part, out);
}